// MLPLowRankPredictor_15676630630927
// MI455X (gfx1250) — compile-verified
//
#include <hip/hip_runtime.h>

// ---------------- problem constants (match reference) ----------------
#define SEQ       64
#define BATCH     32
#define NSAMP     (SEQ * BATCH)   // 2048
#define XD        32
#define ZD        128
#define YD        16
#define HD        128
#define TOTAL_NB  179

// fastfood block ranges (from OFFSETS = [0,32,33,161,162,178,179])
#define L0_WBEG 0
#define L0_WEND 32
#define L0_BIAS 32
#define L1_WBEG 33
#define L1_WEND 161
#define L1_BIAS 161
#define L2_WBEG 162
#define L2_WEND 178
#define L2_BIAS 178

#define SPB   16          // samples per workgroup (= waves per workgroup)
#define NTHR  (SPB * 32)  // 512 threads, 16 wave32 waves
#define HP    132         // padded LDS row stride for [16][128] (bank-conflict free, 16B aligned)
#define XP    36          // padded LDS row stride for [16][32]

typedef __attribute__((ext_vector_type(2))) float v2f;
typedef __attribute__((ext_vector_type(8))) float v8f;

// ---------------- wave-level helpers ----------------

// Unnormalized FWHT over 128 elements laid out as element k = lane + 32*i in u[i].
// FWHT butterfly stages act on disjoint index bits and commute, so order is free:
// bits 0..4 -> cross-lane shfl_xor, bits 5..6 -> in-register.
__device__ __forceinline__ void fwht128(float u[4], int lane) {
#pragma unroll
  for (int d = 1; d <= 16; d <<= 1) {
    const bool hi = (lane & d) != 0;
#pragma unroll
    for (int i = 0; i < 4; ++i) {
      float t = __shfl_xor(u[i], d, 32);
      u[i] = hi ? (t - u[i]) : (u[i] + t);
    }
  }
  {  // distance 32 (register bit 0)
    float a0 = u[0] + u[1], b0 = u[0] - u[1];
    float a1 = u[2] + u[3], b1 = u[2] - u[3];
    u[0] = a0; u[1] = b0; u[2] = a1; u[3] = b1;
  }
  {  // distance 64 (register bit 1)
    float a0 = u[0] + u[2], b0 = u[0] - u[2];
    float a1 = u[1] + u[3], b1 = u[1] - u[3];
    u[0] = a0; u[2] = b0; u[1] = a1; u[3] = b1;
  }
}

__device__ __forceinline__ float waveReduceSum(float x) {
#pragma unroll
  for (int d = 16; d >= 1; d >>= 1) x += __shfl_xor(x, d, 32);
  return x;
}

// One fastfood block r for this wave's sample:
// v = (ffS[r]/128) * FWHT( ffG[r] * perm_r( FWHT( ffB[r] * z ) ) )
// zr[i] holds z[lane + 32*i]; scrat is this wave's private 128-float LDS scratch.
__device__ __forceinline__ void fastfood_block(
    int r, int lane,
    const float zr[4],
    const float* __restrict__ ffB, const float* __restrict__ ffG,
    const float* __restrict__ ffS, const int* __restrict__ ffP,
    float* scrat, float v[4])
{
  const int base = r * 128;
  float u[4];
#pragma unroll
  for (int i = 0; i < 4; ++i)
    u[i] = ffB[base + lane + 32 * i] * zr[i];
  fwht128(u, lane);
  // arbitrary 128-wide permutation via per-wave LDS scratch (in-order DS within a wave)
#pragma unroll
  for (int i = 0; i < 4; ++i) scrat[lane + 32 * i] = u[i];
#pragma unroll
  for (int i = 0; i < 4; ++i) {
    const int k = lane + 32 * i;
    const int idx = ffP[base + k];
    u[i] = ffG[base + k] * scrat[idx];
  }
  fwht128(u, lane);
#pragma unroll
  for (int i = 0; i < 4; ++i)
    v[i] = u[i] * ffS[base + lane + 32 * i] * (1.0f / 128.0f);
}

// ---------------- WMMA tile: C(16x16) = W[m0:m0+16, :K] * H[:K, 0:16] ----------------
// A (16x4 f32): v0 = K=0 (lanes 0-15) | K=2 (lanes 16-31), v1 = K=1 | K=3, M = lane%16.
// B (4x16 f32): mirrored with N = lane%16.
// H is in LDS as [n][k] with row stride ldh.
template <int K>
__device__ __forceinline__ v8f wmma_gemm_tile(const float* __restrict__ W, int m0,
                                              const float* hsrc, int ldh, int lane)
{
  const int hs  = lane >> 4;   // 0: K pair {0,1}, 1: K pair {2,3} of each chunk
  const int l16 = lane & 15;
  v8f acc = {0.f, 0.f, 0.f, 0.f, 0.f, 0.f, 0.f, 0.f};
#pragma unroll
  for (int kc = 0; kc < K; kc += 4) {
    const int col = kc + 2 * hs;
    v2f a, b;
    const float* wr = W + (m0 + l16) * K + col;
    a.x = wr[0];
    a.y = wr[1];
    const float* hr = hsrc + l16 * ldh + col;
    b.x = hr[0];
    b.y = hr[1];
    acc = __builtin_amdgcn_wmma_f32_16x16x4_f32(
        /*neg_a=*/false, a, /*neg_b=*/false, b,
        /*c_mod=*/(short)0, acc, /*reuse_a=*/false, /*reuse_b=*/false);
  }
  return acc;
}

// ---------------- fused kernel: 16 samples per workgroup, 1 wave per sample ----------------
__global__ __launch_bounds__(NTHR)
void mlp_fastfood_fused_kernel(
    const float* __restrict__ x,  const float* __restrict__ z,
    const float* __restrict__ W0, const float* __restrict__ b0,
    const float* __restrict__ W1, const float* __restrict__ b1,
    const float* __restrict__ W2, const float* __restrict__ b2,
    const float* __restrict__ ffB, const float* __restrict__ ffG,
    const float* __restrict__ ffS, const int* __restrict__ ffP,
    float* __restrict__ out)
{
  __shared__ float xlds[SPB * XP];   // inputs as B matrix for layer-0 GEMM
  __shared__ float hA[SPB * HP];     // layer-1 activations
  __shared__ float hB[SPB * HP];     // layer-2 activations
  __shared__ float accs[SPB * HP];   // per-sample correction accumulator (row = own wave)
  __shared__ float perm[SPB * 128];  // per-wave permutation scratch

  const int lane  = threadIdx.x & 31;
  const int wid   = threadIdx.x >> 5;
  const int gbase = blockIdx.x * SPB;
  const int n     = gbase + wid;     // this wave's sample
  const int hs    = lane >> 4;
  const int l16   = lane & 15;

  float* scrat = &perm[wid * 128];

  // per-sample inputs held in registers for the whole kernel
  float zreg[4];
#pragma unroll
  for (int i = 0; i < 4; ++i) zreg[i] = z[n * ZD + lane + 32 * i];
  const float xreg = x[n * XD + lane];
  xlds[wid * XP + lane] = xreg;

  // ---------------- layer 0: corrections (blocks 0..31 = weight rows, 32 = bias) ----------
#pragma unroll 1
  for (int r = L0_WBEG; r < L0_WEND; ++r) {
    __builtin_prefetch(&ffB[(r + 1) * 128 + lane], 0, 0);
    __builtin_prefetch(&ffP[(r + 1) * 128 + lane], 0, 0);
    float v[4];
    fastfood_block(r, lane, zreg, ffB, ffG, ffS, ffP, scrat, v);
    // reg i == weight row 4r+i (32 cols across lanes): corr = dot(row, x[n])
#pragma unroll
    for (int i = 0; i < 4; ++i) {
      const float c = waveReduceSum(v[i] * xreg);
      if (lane == 0) accs[wid * HP + 4 * r + i] = c;
    }
  }
  {  // bias perturbation bd0
    float v[4];
    fastfood_block(L0_BIAS, lane, zreg, ffB, ffG, ffS, ffP, scrat, v);
#pragma unroll
    for (int i = 0; i < 4; ++i) accs[wid * HP + lane + 32 * i] += v[i];
  }
  __syncthreads();

  // ---------------- layer 0 GEMM: h1 = relu(W0*x + acc + b0) -----------------------------
  if (wid < 8) {
    const int m0 = wid * 16;
    v8f d = wmma_gemm_tile<XD>(W0, m0, xlds, XP, lane);
#pragma unroll
    for (int j = 0; j < 8; ++j) {
      const int row = m0 + j + 8 * hs;   // D layout: VGPR j -> M=j (lanes 0-15) / j+8
      const float val = d[j] + accs[l16 * HP + row] + b0[row];
      hA[l16 * HP + row] = fmaxf(val, 0.0f);
    }
  }
  __syncthreads();

  // ---------------- layer 1: corrections (blocks 33..160 = rows, 161 = bias) -------------
  float hreg[4];
#pragma unroll
  for (int i = 0; i < 4; ++i) hreg[i] = hA[wid * HP + lane + 32 * i];
#pragma unroll 1
  for (int r = L1_WBEG; r < L1_WEND; ++r) {
    __builtin_prefetch(&ffB[(r + 1) * 128 + lane], 0, 0);
    __builtin_prefetch(&ffP[(r + 1) * 128 + lane], 0, 0);
    float v[4];
    fastfood_block(r, lane, zreg, ffB, ffG, ffS, ffP, scrat, v);
    float t = v[0] * hreg[0] + v[1] * hreg[1] + v[2] * hreg[2] + v[3] * hreg[3];
    t = waveReduceSum(t);
    if (lane == 0) accs[wid * HP + (r - L1_WBEG)] = t;
  }
  {
    float v[4];
    fastfood_block(L1_BIAS, lane, zreg, ffB, ffG, ffS, ffP, scrat, v);
#pragma unroll
    for (int i = 0; i < 4; ++i) accs[wid * HP + lane + 32 * i] += v[i];
  }
  __syncthreads();

  // ---------------- layer 1 GEMM: h2 = relu(W1*h1 + acc + b1) ----------------------------
  if (wid < 8) {
    const int m0 = wid * 16;
    v8f d = wmma_gemm_tile<HD>(W1, m0, hA, HP, lane);
#pragma unroll
    for (int j = 0; j < 8; ++j) {
      const int row = m0 + j + 8 * hs;
      const float val = d[j] + accs[l16 * HP + row] + b1[row];
      hB[l16 * HP + row] = fmaxf(val, 0.0f);
    }
  }
  __syncthreads();

  // ---------------- layer 2: corrections (blocks 162..177 = rows, 178 = bias[0:16]) ------
  float h2reg[4];
#pragma unroll
  for (int i = 0; i < 4; ++i) h2reg[i] = hB[wid * HP + lane + 32 * i];
#pragma unroll 1
  for (int r = L2_WBEG; r < L2_WEND; ++r) {
    float v[4];
    fastfood_block(r, lane, zreg, ffB, ffG, ffS, ffP, scrat, v);
    float t = v[0] * h2reg[0] + v[1] * h2reg[1] + v[2] * h2reg[2] + v[3] * h2reg[3];
    t = waveReduceSum(t);
    if (lane == 0) accs[wid * HP + (r - L2_WBEG)] = t;
  }
  {
    float v[4];
    fastfood_block(L2_BIAS, lane, zreg, ffB, ffG, ffS, ffP, scrat, v);
    if (lane < 16) accs[wid * HP + lane] += v[0];  // only first 16 of the block are bd2
  }
  __syncthreads();

  // ---------------- layer 2 GEMM: y = W2*h2 + acc + b2 -> global out ---------------------
  if (wid == 0) {
    v8f d = wmma_gemm_tile<HD>(W2, 0, hB, HP, lane);
#pragma unroll
    for (int j = 0; j < 8; ++j) {
      const int row = j + 8 * hs;                         // output feature 0..15
      out[(gbase + l16) * YD + row] = d[j] + accs[l16 * HP + row] + b2[row];
    }
  }
}

// ---------------- host entry ----------------
extern "C" void kernel_launch(void* const* d_in, const int* in_sizes, int n_in,
                              void* d_out, int out_size, void* d_ws, size_t ws_size,
                              hipStream_t stream) {
  (void)in_sizes; (void)n_in; (void)out_size; (void)d_ws; (void)ws_size;
  const float* x   = (const float*)d_in[0];
  const float* z   = (const float*)d_in[1];
  const float* W0  = (const float*)d_in[2];
  const float* b0  = (const float*)d_in[3];
  const float* W1  = (const float*)d_in[4];
  const float* b1  = (const float*)d_in[5];
  const float* W2  = (const float*)d_in[6];
  const float* b2  = (const float*)d_in[7];
  const float* ffB = (const float*)d_in[8];
  const float* ffG = (const float*)d_in[9];
  const float* ffS = (const float*)d_in[10];
  const int*   ffP = (const int*)d_in[11];
  float* out = (float*)d_out;

  dim3 grid(NSAMP / SPB);   // 128 workgroups
  dim3 block(NTHR);         // 512 threads = 16 wave32 waves
  mlp_fastfood_fused_kernel<<<grid, block, 0, stream>>>(
      x, z, W0, b0, W1, b1, W2, b2, ffB, ffG, ffS, ffP, out);
}